// UPP_Layer_55559696941260
// MI455X (gfx1250) — compile-verified
//
#include <hip/hip_runtime.h>
#include <cstdint>

typedef __attribute__((ext_vector_type(2))) float v2f;
typedef __attribute__((ext_vector_type(4))) float v4f;
typedef __attribute__((ext_vector_type(8))) float v8f;

// One 16-byte async global->LDS copy per lane (gfx1250 VGLOBAL async op, ASYNCcnt).
#define ASYNC_CP16(ldsOff, gPtr)                                            \
  asm volatile("global_load_async_to_lds_b128 %0, %1, off"                  \
               :: "v"(ldsOff), "v"(gPtr) : "memory")

__global__ __launch_bounds__(64) void pool7x7_wmma(const float* __restrict__ x,
                                                   float* __restrict__ out) {
  constexpr int C    = 2048;
  constexpr int HW   = 784;      // 28*28
  constexpr int ROWF = 116;      // 112 data floats + 4 pad (bank decorrelation)
  constexpr int BUFF = 32 * ROWF;

  __shared__ float smem[2 /*waves*/ * 2 /*dbl buf*/ * BUFF];   // 59,392 B

  const int tid  = threadIdx.x;
  const int wave = tid >> 5;
  const int lane = tid & 31;
  const int b    = blockIdx.x >> 5;          // 32 c-blocks of 64 channels per batch
  const int cblk = blockIdx.x & 31;
  const int cw   = cblk * 64 + wave * 32;    // this wave's 32-channel base
  const int myc  = cw + lane;                // plane this lane stages

  const float* gplane = x + ((size_t)b * C + myc) * HW;

  // Low 32 bits of the generic address of a __shared__ object == LDS byte offset.
  const unsigned smemByte  = (unsigned)(uintptr_t)(&smem[0]);
  const unsigned waveByte  = smemByte + (unsigned)(wave * (2 * BUFF) * 4);
  const unsigned laneByte0 = waveByte + (unsigned)(lane * ROWF * 4);
  const unsigned laneByte1 = laneByte0 + (unsigned)(BUFF * 4);

  // Stage i-group ig (4 rows = 112 floats = 28 x 16B) into buffer s.
  auto issue = [&](int ig, int s) {
    const float* g = gplane + ig * 112;
    const unsigned l = s ? laneByte1 : laneByte0;
#pragma unroll
    for (int r = 0; r < 28; ++r) {
      ASYNC_CP16(l + (unsigned)(r * 16), g + r * 4);
    }
  };

  issue(0, 0);

  const int m  = lane & 15;                 // WMMA M / B-column row index
  const int tb = (lane < 16) ? 0 : 2;       // A/B K-sublane base per ISA layout

#pragma unroll
  for (int i = 0; i < 7; ++i) {
    if (i < 6) {
      issue(i + 1, (i + 1) & 1);
      asm volatile("s_wait_asynccnt 28" ::: "memory");  // group i complete
    } else {
      asm volatile("s_wait_asynccnt 0" ::: "memory");   // last group complete
    }

    const float* bufp = &smem[wave * (2 * BUFF) + (i & 1) * BUFF];

    v8f D0 = {0.f, 0.f, 0.f, 0.f, 0.f, 0.f, 0.f, 0.f};  // channels cw+0..15
    v8f D1 = {0.f, 0.f, 0.f, 0.f, 0.f, 0.f, 0.f, 0.f};  // channels cw+16..31

#pragma unroll
    for (int kb = 0; kb < 7; ++kb) {   // K-step kb <-> output column-bin j = kb
      const float av = (m == kb) ? 0.0625f : 0.0f;      // Ph*Pw weight 1/16
      v2f a; a.x = av; a.y = av;                        // K=v,v+1 share value

      {  // B fragment, channel-half 0: lane reads plane (cw + m)
        const float* s0 = bufp + (0 * 16 + m) * ROWF + tb * 28 + kb * 4;
        const v4f q0 = *(const v4f*)(s0);
        const v4f q1 = *(const v4f*)(s0 + 28);
        v2f bb;
        bb.x = (q0.x + q0.y) + (q0.z + q0.w);
        bb.y = (q1.x + q1.y) + (q1.z + q1.w);
        D0 = __builtin_amdgcn_wmma_f32_16x16x4_f32(false, a, false, bb,
                                                   (short)0, D0, false, false);
      }
      {  // B fragment, channel-half 1: lane reads plane (cw + 16 + m)
        const float* s1 = bufp + (16 + m) * ROWF + tb * 28 + kb * 4;
        const v4f q0 = *(const v4f*)(s1);
        const v4f q1 = *(const v4f*)(s1 + 28);
        v2f bb;
        bb.x = (q0.x + q0.y) + (q0.z + q0.w);
        bb.y = (q1.x + q1.y) + (q1.z + q1.w);
        D1 = __builtin_amdgcn_wmma_f32_16x16x4_f32(false, a, false, bb,
                                                   (short)0, D1, false, false);
      }
    }

    // D rows 0..6 (= j) live in lanes 0..15, VGPRs 0..6. Store coalesced 64B runs.
    if (lane < 16) {
      const size_t base = ((size_t)b * 49 + (size_t)i * 7) * C + (size_t)cw;
#pragma unroll
      for (int d = 0; d < 7; ++d) {
        out[base + (size_t)d * C + lane]      = D0[d];
        out[base + (size_t)d * C + 16 + lane] = D1[d];
      }
    }
  }
}

extern "C" void kernel_launch(void* const* d_in, const int* in_sizes, int n_in,
                              void* d_out, int out_size, void* d_ws, size_t ws_size,
                              hipStream_t stream) {
  const float* x  = (const float*)d_in[0];
  float* out      = (float*)d_out;
  // 32 batches * 32 channel-blocks of 64 = 1024 blocks, 64 threads (2 waves).
  dim3 grid(1024), block(64);
  hipLaunchKernelGGL(pool7x7_wmma, grid, block, 0, stream, x, out);
}